// CausalMultiHeadSelfAttention_1511828488590
// MI455X (gfx1250) — compile-verified
//
#include <hip/hip_runtime.h>
#include <math.h>

typedef __attribute__((ext_vector_type(2))) float v2f;
typedef __attribute__((ext_vector_type(8))) float v8f;

#define D_MODEL   1024
#define NUM_HEADS 16
#define HEAD_DIM  64
#define Bsz       2
#define Tseq      2048
#define MROWS     (Bsz * Tseq)          // 4096
#define QTILES    (Tseq / 16)           // 128

// ---------------------------------------------------------------------------
// fp32 WMMA: D(16x16,f32) = A(16x4,f32) * B(4x16,f32) + C
// A/B fragments: 2 VGPRs/lane. lanes 0-15 hold K=0,1 ; lanes 16-31 hold K=2,3.
// C/D: VGPR r, lane l -> (row r + 8*(l>>4), col l&15).
// ---------------------------------------------------------------------------
__device__ __forceinline__ v8f wmma4(v2f a, v2f b, v8f c) {
  return __builtin_amdgcn_wmma_f32_16x16x4_f32(false, a, false, b,
                                               (short)0, c, false, false);
}

__device__ __forceinline__ float rowmax16(float v) {
  v = fmaxf(v, __shfl_xor(v, 1, 32));
  v = fmaxf(v, __shfl_xor(v, 2, 32));
  v = fmaxf(v, __shfl_xor(v, 4, 32));
  v = fmaxf(v, __shfl_xor(v, 8, 32));
  return v;
}
__device__ __forceinline__ float rowsum16(float v) {
  v += __shfl_xor(v, 1, 32);
  v += __shfl_xor(v, 2, 32);
  v += __shfl_xor(v, 4, 32);
  v += __shfl_xor(v, 8, 32);
  return v;
}

// 16x64 output tile of X(MROWSxK row-major) @ W(KxN row-major), n0 = 64-aligned
__device__ __forceinline__ void gemm16x64(const float* __restrict__ X,
                                          const float* __restrict__ W,
                                          int m0, int n0, int lane, v8f acc[4]) {
  const int half = lane >> 4, lrow = lane & 15;
  const float* ap = X + (m0 + lrow) * D_MODEL + 2 * half;
  const float* wp = W + 2 * half * D_MODEL + n0 + lrow;
#pragma unroll 4
  for (int k0 = 0; k0 < D_MODEL; k0 += 4) {
    v2f a = *(const v2f*)(ap + k0);
    const float* wk = wp + k0 * D_MODEL;
#pragma unroll
    for (int t = 0; t < 4; ++t) {
      v2f b;
      b.x = wk[16 * t];
      b.y = wk[16 * t + D_MODEL];
      acc[t] = wmma4(a, b, acc[t]);
    }
  }
}

// ---------------------------------------------------------------------------
// Zero-fill (b128 stores) for the 536MB attn output region
// ---------------------------------------------------------------------------
__global__ __launch_bounds__(256) void fill_zero_kernel(float4* __restrict__ p,
                                                        size_t n4) {
  size_t i = (size_t)blockIdx.x * blockDim.x + threadIdx.x;
  size_t stride = (size_t)gridDim.x * blockDim.x;
  float4 z = make_float4(0.f, 0.f, 0.f, 0.f);
  for (; i < n4; i += stride) p[i] = z;
}

// ---------------------------------------------------------------------------
// Q/K/V projections: out[(b,h,t,d)] = (x @ W + b) split into heads
// grid (64, 16, 3), block 128 (4 waves, each a 16-row tile)
// ---------------------------------------------------------------------------
__global__ __launch_bounds__(128) void qkv_kernel(
    const float* __restrict__ x,
    const float* __restrict__ Wq, const float* __restrict__ bq,
    const float* __restrict__ Wk, const float* __restrict__ bk,
    const float* __restrict__ Wv, const float* __restrict__ bv,
    float* __restrict__ Q, float* __restrict__ K, float* __restrict__ V) {
  const int w = threadIdx.x >> 5, lane = threadIdx.x & 31;
  const int half = lane >> 4, lrow = lane & 15;
  const int m0 = (blockIdx.x * 4 + w) * 16;
  const int h = blockIdx.y;
  const int n0 = h * HEAD_DIM;

  const float* W;
  const float* bias;
  float* out;
  if (blockIdx.z == 0)      { W = Wq; bias = bq; out = Q; }
  else if (blockIdx.z == 1) { W = Wk; bias = bk; out = K; }
  else                      { W = Wv; bias = bv; out = V; }

  v8f z8 = {0.f, 0.f, 0.f, 0.f, 0.f, 0.f, 0.f, 0.f};
  v8f acc[4] = {z8, z8, z8, z8};
  gemm16x64(x, W, m0, n0, lane, acc);

  const int bb = m0 / Tseq;
  const int tok0 = m0 % Tseq;
  float* oh = out + ((size_t)(bb * NUM_HEADS + h) * Tseq) * HEAD_DIM;
#pragma unroll
  for (int t = 0; t < 4; ++t) {
    const float bn = bias[n0 + 16 * t + lrow];
    const int d = 16 * t + lrow;
#pragma unroll
    for (int r = 0; r < 8; ++r) {
      const int tok = tok0 + r + 8 * half;
      oh[(size_t)tok * HEAD_DIM + d] = acc[t][r] + bn;
    }
  }
}

// ---------------------------------------------------------------------------
// Attention: one wave per (b, h, 16-query tile). Two-pass flash softmax.
// Allowed keys for query i: j in [max(0,128*(blk-1)), i]  (window/block proven
// equivalent to this range), so <=17 key tiles of 16 per query tile.
// ---------------------------------------------------------------------------
__global__ __launch_bounds__(128) void attn_kernel(
    const float* __restrict__ Q, const float* __restrict__ K,
    const float* __restrict__ V, float* __restrict__ attnOut,
    float* __restrict__ Y) {
  __shared__ __align__(16) float psh[4][16 * 18];  // padded P tile per wave

  const int w = threadIdx.x >> 5, lane = threadIdx.x & 31;
  const int half = lane >> 4, lrow = lane & 15;
  const int tile = blockIdx.x * 4 + w;
  const int b = tile / (NUM_HEADS * QTILES);
  const int rem = tile % (NUM_HEADS * QTILES);
  const int h = rem / QTILES;
  const int i0 = (rem % QTILES) * 16;

  const size_t hoff = (size_t)(b * NUM_HEADS + h) * Tseq;
  const float* Qh = Q + hoff * HEAD_DIM;
  const float* Kh = K + hoff * HEAD_DIM;
  const float* Vh = V + hoff * HEAD_DIM;
  float* arow = attnOut + hoff * Tseq;

  const int blk = i0 >> 7;
  const int jstart = (blk > 0) ? (blk - 1) * 128 : 0;
  const int jend = i0 + 16;  // exclusive
  const float slope = exp2f(-0.5f * (float)(h + 1));
  const float sscale = 0.125f;  // 1/sqrt(64)
  const float NEGBIG = -3.0e38f;

  const float* qp = Qh + (i0 + lrow) * HEAD_DIM + 2 * half;

  float mrow[8], ssum[8];
#pragma unroll
  for (int r = 0; r < 8; ++r) { mrow[r] = NEGBIG; ssum[r] = 0.f; }

  v8f z8 = {0.f, 0.f, 0.f, 0.f, 0.f, 0.f, 0.f, 0.f};

  // -------- pass 1: softmax stats --------
  for (int j0 = jstart; j0 < jend; j0 += 16) {
    v8f acc = z8;
    const float* kp = Kh + (j0 + lrow) * HEAD_DIM + 2 * half;
#pragma unroll
    for (int d0 = 0; d0 < HEAD_DIM; d0 += 4) {
      v2f a = *(const v2f*)(qp + d0);
      v2f bf = *(const v2f*)(kp + d0);
      acc = wmma4(a, bf, acc);
    }
    const int kj = j0 + lrow;
#pragma unroll
    for (int r = 0; r < 8; ++r) {
      const int qi = i0 + r + 8 * half;
      float s = acc[r] * sscale + slope * (float)(qi - kj);
      if (kj > qi) s = NEGBIG;
      const float nm = fmaxf(mrow[r], rowmax16(s));
      const float rs = rowsum16(__expf(s - nm));
      ssum[r] = ssum[r] * __expf(mrow[r] - nm) + rs;
      mrow[r] = nm;
    }
  }
  float rinv[8];
#pragma unroll
  for (int r = 0; r < 8; ++r) rinv[r] = 1.0f / ssum[r];

  // -------- pass 2: probabilities + P@V --------
  v8f oacc[4] = {z8, z8, z8, z8};
  float* pw = &psh[w][0];
  for (int j0 = jstart; j0 < jend; j0 += 16) {
    v8f acc = z8;
    const float* kp = Kh + (j0 + lrow) * HEAD_DIM + 2 * half;
#pragma unroll
    for (int d0 = 0; d0 < HEAD_DIM; d0 += 4) {
      v2f a = *(const v2f*)(qp + d0);
      v2f bf = *(const v2f*)(kp + d0);
      acc = wmma4(a, bf, acc);
    }
    const int kj = j0 + lrow;
#pragma unroll
    for (int r = 0; r < 8; ++r) {
      const int qi = i0 + r + 8 * half;
      float s = acc[r] * sscale + slope * (float)(qi - kj);
      if (kj > qi) s = NEGBIG;
      const float p = __expf(s - mrow[r]) * rinv[r];
      arow[(size_t)qi * Tseq + kj] = p;          // band write to d_out
      pw[(r + 8 * half) * 18 + lrow] = p;        // stage for transpose
    }
    asm volatile("s_wait_dscnt 0" ::: "memory"); // LDS stores visible to loads
#pragma unroll
    for (int s4 = 0; s4 < 4; ++s4) {
      v2f a = *(const v2f*)(pw + lrow * 18 + 4 * s4 + 2 * half);  // A-frag of P
      const float* vp = Vh + (j0 + 4 * s4 + 2 * half) * HEAD_DIM + lrow;
#pragma unroll
      for (int t = 0; t < 4; ++t) {
        v2f bf;
        bf.x = vp[16 * t];
        bf.y = vp[16 * t + HEAD_DIM];
        oacc[t] = wmma4(a, bf, oacc[t]);
      }
    }
    asm volatile("s_wait_dscnt 0" ::: "memory"); // WAR before psh reuse
  }

  // write O back in (B,T,C) layout for the output projection
  const int m0 = b * Tseq + i0;
#pragma unroll
  for (int t = 0; t < 4; ++t) {
#pragma unroll
    for (int r = 0; r < 8; ++r) {
      const int m = m0 + r + 8 * half;
      Y[(size_t)m * D_MODEL + h * HEAD_DIM + 16 * t + lrow] = oacc[t][r];
    }
  }
}

// ---------------------------------------------------------------------------
// Output projection: out = Y @ W_o + b_o   (writes first section of d_out)
// ---------------------------------------------------------------------------
__global__ __launch_bounds__(128) void oproj_kernel(
    const float* __restrict__ Y, const float* __restrict__ Wo,
    const float* __restrict__ bo, float* __restrict__ out) {
  const int w = threadIdx.x >> 5, lane = threadIdx.x & 31;
  const int half = lane >> 4, lrow = lane & 15;
  const int m0 = (blockIdx.x * 4 + w) * 16;
  const int n0 = blockIdx.y * 64;

  v8f z8 = {0.f, 0.f, 0.f, 0.f, 0.f, 0.f, 0.f, 0.f};
  v8f acc[4] = {z8, z8, z8, z8};
  gemm16x64(Y, Wo, m0, n0, lane, acc);

#pragma unroll
  for (int t = 0; t < 4; ++t) {
    const float bn = bo[n0 + 16 * t + lrow];
#pragma unroll
    for (int r = 0; r < 8; ++r) {
      const int m = m0 + r + 8 * half;
      out[(size_t)m * D_MODEL + n0 + 16 * t + lrow] = acc[t][r] + bn;
    }
  }
}

// ---------------------------------------------------------------------------
extern "C" void kernel_launch(void* const* d_in, const int* in_sizes, int n_in,
                              void* d_out, int out_size, void* d_ws,
                              size_t ws_size, hipStream_t stream) {
  const float* x  = (const float*)d_in[0];
  const float* Wq = (const float*)d_in[1];
  const float* bq = (const float*)d_in[2];
  const float* Wk = (const float*)d_in[3];
  const float* bk = (const float*)d_in[4];
  const float* Wv = (const float*)d_in[5];
  const float* bv = (const float*)d_in[6];
  const float* Wo = (const float*)d_in[7];
  const float* bo = (const float*)d_in[8];

  float* out = (float*)d_out;                                  // (B,T,C)
  float* attn = out + (size_t)Bsz * Tseq * D_MODEL;            // (B,H,T,T)

  const size_t mat = (size_t)MROWS * D_MODEL;  // 4096*1024 floats
  float* Q = (float*)d_ws;
  float* K = Q + mat;
  float* V = K + mat;
  float* Y = V + mat;

  // 1) zero the attention output (dominant HBM cost: 536 MB of stores)
  const size_t n4 = ((size_t)Bsz * NUM_HEADS * Tseq * Tseq) / 4;
  fill_zero_kernel<<<8192, 256, 0, stream>>>((float4*)attn, n4);

  // 2) Q/K/V projections via fp32 WMMA
  qkv_kernel<<<dim3(MROWS / 64, NUM_HEADS, 3), 128, 0, stream>>>(
      x, Wq, bq, Wk, bk, Wv, bv, Q, K, V);

  // 3) attention (4096 query tiles, 4 waves per block)
  attn_kernel<<<dim3(Bsz * NUM_HEADS * QTILES / 4), 128, 0, stream>>>(
      Q, K, V, attn, Y);

  // 4) output projection
  oproj_kernel<<<dim3(MROWS / 64, D_MODEL / 64), 128, 0, stream>>>(Y, Wo, bo,
                                                                   out);
}